// PoseProjection_40114994545037
// MI455X (gfx1250) — compile-verified
//
#include <hip/hip_runtime.h>
#include <hip/hip_bf16.h>

// ---------------------------------------------------------------------------
// PoseProjection on MI455X (gfx1250, wave32)
//
// Outputs concatenated flat in return order (all float32):
//   [0]            proj_features   N*64
//   [N*64]         proj_sdf        N
//   [N*65]         proj_occupancy  N
//   [N*66]         historical_crd  N*3
//   [N*69]         mask            N   (1.0f / 0.0f)
//
// Kernel 1 (1 block x 32 threads): compose T_b = inv(C_b) @ H_b for the 8
//   batches using two V_WMMA_F32_16X16X4_F32 ops (4 batches per WMMA, packed
//   block-diagonally). Writes 8 x 12 floats (row-major 3x4) into d_ws.
// Kernel 2 (N/256 blocks x 256 threads): streaming transform + masked copy,
//   fully coalesced b128 traffic on the dominant 1 GB features stream.
// ---------------------------------------------------------------------------

typedef __attribute__((ext_vector_type(2))) float v2f;
typedef __attribute__((ext_vector_type(8))) float v8f;

// Compose 4 batches' transforms with one WMMA.
//   A (16x4): row r = invRigid(C_{p4+(r>>2)}) row (r&3)
//   B (4x16): col c = H_{p4+(c>>2)} column (c&3)
//   D (16x16): diagonal 4x4 blocks b hold T_b = invC_b @ H_b
// VGPR layouts per CDNA5 ISA 7.12.2 (32-bit, wave32):
//   A: lanes 0-15 rows 0-15 with K=0,1 in VGPR0,1; lanes 16-31 same rows K=2,3
//   B: lane = column (mod 16); lanes 0-15 K=0,1; lanes 16-31 K=2,3
//   D: lane L holds column L%16, rows M = 8*(L>=16) + v for VGPR v=0..7
__device__ __forceinline__ void compose4(const float* __restrict__ hist,
                                         const float* __restrict__ cur,
                                         float* __restrict__ T_out,
                                         int lane, int p4) {
  const int rc   = lane & 15;        // A row / B column index (0..15)
  const int k0   = (lane >> 4) * 2;  // this lane supplies K = k0, k0+1
  const int b    = p4 + (rc >> 2);   // batch for this lane's A row / B col
  const int ij   = rc & 3;           // row-within-transform (A) / col (B)

  // --- A operand: inverse of rigid pose C_b, row ij, elements k0, k0+1.
  // inv([[R,t],[0,1]]) = [[R^T, -R^T t],[0,0,0,1]]   (row-major 4x4 in memory)
  const float* C = cur + b * 16;
  float a[2];
  #pragma unroll
  for (int q = 0; q < 2; ++q) {
    int k = k0 + q;
    float val;
    if (ij == 3) {
      val = (k == 3) ? 1.0f : 0.0f;                 // bottom row [0,0,0,1]
    } else if (k < 3) {
      val = C[k * 4 + ij];                          // R^T[ij][k] = R[k][ij]
    } else {
      val = -(C[0 * 4 + ij] * C[3] +                // -(R^T t)[ij]
              C[1 * 4 + ij] * C[7] +
              C[2 * 4 + ij] * C[11]);
    }
    a[q] = val;
  }

  // --- B operand: H_b column ij, rows k0, k0+1.
  const float* H = hist + b * 16;
  v2f A = { a[0], a[1] };
  v2f B = { H[k0 * 4 + ij], H[(k0 + 1) * 4 + ij] };
  v8f Cacc = {};

  // 8-arg form: (neg_a, A, neg_b, B, c_mod, C, reuse_a, reuse_b)
  v8f D = __builtin_amdgcn_wmma_f32_16x16x4_f32(
      false, A, false, B, (short)0, Cacc, false, false);

  // --- Scatter the block-diagonal 4x4 results: this lane holds column
  // col = rc, rows M = mbase + v. Keep rows r with r>>2 == col>>2 and
  // (r&3) < 3 (we only need the top 3 rows of each 3x4 transform).
  const int col   = rc;
  const int mbase = (lane >> 4) * 8;
  #pragma unroll
  for (int v = 0; v < 8; ++v) {
    int r  = mbase + v;
    int br = r >> 2;
    int ir = r & 3;
    if (ir < 3 && (col >> 2) == br) {
      T_out[(p4 + br) * 12 + ir * 4 + (col & 3)] = D[v];
    }
  }
}

__global__ __launch_bounds__(32)
void pose_compose_wmma(const float* __restrict__ hist,
                       const float* __restrict__ cur,
                       float* __restrict__ T_out) {
  const int lane = threadIdx.x;          // one wave, EXEC all-ones for WMMA
  compose4(hist, cur, T_out, lane, 0);   // batches 0..3
  compose4(hist, cur, T_out, lane, 4);   // batches 4..7
}

#define VOX_PER_BLOCK 256

__global__ __launch_bounds__(VOX_PER_BLOCK)
void pose_project(const float* __restrict__ coords,
                  const int*   __restrict__ binds,
                  const float* __restrict__ features,
                  const float* __restrict__ sdf,
                  const float* __restrict__ occ,
                  const float* __restrict__ Ttab,   // 8 x 12 floats
                  float* __restrict__ out,
                  int n) {
  __shared__ float sT[96];
  __shared__ float smask[VOX_PER_BLOCK];

  const int t = threadIdx.x;
  if (t < 96) sT[t] = Ttab[t];
  __syncthreads();

  const long long v0 = (long long)blockIdx.x * VOX_PER_BLOCK;
  const long long v  = v0 + t;
  const long long N  = n;

  float* __restrict__ out_feat = out;
  float* __restrict__ out_sdf  = out + N * 64;
  float* __restrict__ out_occ  = out + N * 65;
  float* __restrict__ out_hc   = out + N * 66;
  float* __restrict__ out_mask = out + N * 69;

  float m = 0.0f;
  if (v < N) {
    const int b = binds[v];
    const float* __restrict__ T = sT + b * 12;
    const float x = coords[v * 3 + 0];
    const float y = coords[v * 3 + 1];
    const float z = coords[v * 3 + 2];

    const float hx = fmaf(T[0], x, fmaf(T[1], y, fmaf(T[2],  z, T[3])));
    const float hy = fmaf(T[4], x, fmaf(T[5], y, fmaf(T[6],  z, T[7])));
    const float hz = fmaf(T[8], x, fmaf(T[9], y, fmaf(T[10], z, T[11])));

    // vox = h / 0.0625 == h * 16 (exact)
    const float vx = hx * 16.0f, vy = hy * 16.0f, vz = hz * 16.0f;
    const bool inb = (vx >= 0.0f) && (vx < 96.0f) &&
                     (vy >= 0.0f) && (vy < 96.0f) &&
                     (vz >= 0.0f) && (vz < 48.0f);
    m = inb ? 1.0f : 0.0f;

    out_hc[v * 3 + 0] = hx;
    out_hc[v * 3 + 1] = hy;
    out_hc[v * 3 + 2] = hz;
    out_sdf[v]  = m * sdf[v];
    out_occ[v]  = m * occ[v];
    out_mask[v] = m;
  }
  smask[t] = m;
  __syncthreads();

  // Features: this block covers 256 voxels x 64 ch = 4096 float4 chunks.
  // Iteration k: lane t moves chunk f = k*256 + t -> consecutive lanes hit
  // consecutive 16B chunks (global_load/store_b128, 4 KB per iteration).
  const float4* __restrict__ fin  = (const float4*)features + v0 * 16;
  float4*       __restrict__ fout = (float4*)out_feat       + v0 * 16;
  #pragma unroll
  for (int k = 0; k < 16; ++k) {
    const int f  = k * VOX_PER_BLOCK + t;   // 0..4095
    const int vl = f >> 4;                  // local voxel 0..255
    if (v0 + vl < N) {
      const float mm = smask[vl];
      float4 d = fin[f];
      d.x *= mm; d.y *= mm; d.z *= mm; d.w *= mm;
      fout[f] = d;
    }
  }
}

extern "C" void kernel_launch(void* const* d_in, const int* in_sizes, int n_in,
                              void* d_out, int out_size, void* d_ws, size_t ws_size,
                              hipStream_t stream) {
  const float* coords   = (const float*)d_in[0];
  const int*   binds    = (const int*)  d_in[1];
  const float* features = (const float*)d_in[2];
  const float* sdf      = (const float*)d_in[3];
  const float* occ      = (const float*)d_in[4];
  const float* hist     = (const float*)d_in[5];
  const float* cur      = (const float*)d_in[6];
  float* out  = (float*)d_out;
  float* Ttab = (float*)d_ws;          // 8 * 12 floats = 384 B scratch

  const int n = in_sizes[1];           // batch_inds count == N_VOXELS

  pose_compose_wmma<<<1, 32, 0, stream>>>(hist, cur, Ttab);

  const int blocks = (n + VOX_PER_BLOCK - 1) / VOX_PER_BLOCK;
  pose_project<<<blocks, VOX_PER_BLOCK, 0, stream>>>(
      coords, binds, features, sdf, occ, Ttab, out, n);
}